// Postprocess_19928648253798
// MI455X (gfx1250) — compile-verified
//
#include <hip/hip_runtime.h>
#include <stdint.h>

// Problem constants (reference: 8192x8192 fp32, 9x9 maxpool, thresh 0.5).
#define IMG_H 8192
#define IMG_W 8192
#define RAD   4
#define TILE  64
#define IN_DIM (TILE + 2 * RAD)   // 72: tile + halo
#define NTHREADS 256

typedef __attribute__((ext_vector_type(4))) uint32_t u32x4;
typedef __attribute__((ext_vector_type(8))) uint32_t u32x8;
typedef __attribute__((ext_vector_type(4))) float    f32x4;

__device__ __forceinline__ f32x4 vmax4(f32x4 a, f32x4 b) {
  f32x4 r;
  r.x = fmaxf(a.x, b.x);
  r.y = fmaxf(a.y, b.y);
  r.z = fmaxf(a.z, b.z);
  r.w = fmaxf(a.w, b.w);
  return r;
}

// ---------------------------------------------------------------------------
// gfx1250 async DMA: global -> LDS per-lane b32, tracked by ASYNCcnt.
// Used only by border blocks (needs per-element clamp-to-edge addressing).
// ---------------------------------------------------------------------------
__device__ __forceinline__ void async_load_b32_to_lds(uint32_t lds_byte_off,
                                                      const float* gaddr) {
  asm volatile("global_load_async_to_lds_b32 %0, %1, off"
               :
               : "v"(lds_byte_off), "v"(gaddr)
               : "memory");
}

__device__ __forceinline__ void wait_asynccnt0() {
  asm volatile("s_wait_asynccnt 0" ::: "memory");
}

// ---------------------------------------------------------------------------
// gfx1250 Tensor Data Mover: one DMA moves the whole 72x72 fp32 tile
// (strided 2D rectangle, row stride 8192 elements) into contiguous LDS.
// D# groups per CDNA5 ISA; VADDR2/3 = NULL (2D).
// Tracked by TENSORcnt; EXEC ignored, so only one wave may issue it.
// ---------------------------------------------------------------------------
__device__ __forceinline__ void tdm_load_tile_2d(uint32_t lds_addr,
                                                 uint64_t global_addr) {
  u32x4 g0;
  g0.x = 1u;                                         // count=1 (valid D#)
  g0.y = lds_addr;                                   // LDS dest (bytes)
  g0.z = (uint32_t)global_addr;                      // global_addr[31:0]
  g0.w = (uint32_t)(global_addr >> 32) | (2u << 30); // addr[56:32] | type=2

  u32x8 g1;
  g1.s0 = 2u << 16;                       // data_size=2 -> 4-byte elements
  g1.s1 = ((uint32_t)IMG_W & 0xFFFFu) << 16;          // tensor_dim0 lo16
  g1.s2 = ((uint32_t)IMG_W >> 16) |                   // tensor_dim0 hi16
          (((uint32_t)IMG_H & 0xFFFFu) << 16);        // tensor_dim1 lo16
  g1.s3 = ((uint32_t)IMG_H >> 16) |                   // tensor_dim1 hi16
          ((uint32_t)IN_DIM << 16);                   // tile_dim0 = 72
  g1.s4 = (uint32_t)IN_DIM;               // tile_dim1 = 72, tile_dim2 = 0 (2D)
  g1.s5 = (uint32_t)IMG_W;                // tensor_dim0_stride lo32 = 8192
  g1.s6 = 0u;                             // stride hi16 | dim1_stride lo16
  g1.s7 = 0u;                             // dim1_stride hi32 (unused, 2D)

  asm volatile("tensor_load_to_lds %0, %1"
               :
               : "s"(g0), "s"(g1)
               : "memory");
}

// Low 32 bits of a generic pointer to a __shared__ object are the LDS offset
// (flat LDS aperture: offset is the low 32 address bits).
__device__ __forceinline__ uint32_t lds_offset_of(const void* p) {
  return (uint32_t)(uintptr_t)p;
}

__global__ __launch_bounds__(NTHREADS)
void peak_pool_kernel(const float* __restrict__ in, float* __restrict__ out) {
  // 72*72*4 = 20736 B raw tile, 72*64*4 = 18432 B horizontal-max buffer.
  __shared__ float tile[IN_DIM * IN_DIM];
  __shared__ float hmax[IN_DIM * TILE];

  const int tid = threadIdx.x;
  const int bx = blockIdx.x;
  const int by = blockIdx.y;
  const int x0 = bx * TILE - RAD;
  const int y0 = by * TILE - RAD;

  const uint32_t tile_base = lds_offset_of(&tile[0]);

  const bool interior = (x0 >= 0) & (y0 >= 0) &
                        (x0 + IN_DIM <= IMG_W) & (y0 + IN_DIM <= IMG_H);

  if (interior) {
    // ---- One TDM descriptor moves the whole halo tile. Wave 0 only.
    if (tid < 32) {
      uint64_t gaddr =
          (uint64_t)(uintptr_t)(in + (size_t)y0 * IMG_W + (size_t)x0);
      tdm_load_tile_2d(tile_base, gaddr);
      __builtin_amdgcn_s_wait_tensorcnt(0);
    }
  } else {
    // ---- Border blocks: per-lane async b32 with clamp-to-edge (== -inf pad
    // for max pooling, since max over duplicated in-range values is the same).
    constexpr int NLOAD = IN_DIM * IN_DIM;  // 5184
    for (int idx = tid; idx < NLOAD; idx += NTHREADS) {
      int r = idx / IN_DIM;
      int c = idx - r * IN_DIM;
      int gy = y0 + r;
      gy = gy < 0 ? 0 : (gy > IMG_H - 1 ? IMG_H - 1 : gy);
      int gx = x0 + c;
      gx = gx < 0 ? 0 : (gx > IMG_W - 1 ? IMG_W - 1 : gx);
      const float* g = in + (size_t)gy * IMG_W + gx;
      async_load_b32_to_lds(tile_base + 4u * (uint32_t)idx, g);
    }
    wait_asynccnt0();  // each wave drains its own ASYNCcnt
  }
  __syncthreads();     // tile visible to all waves

  // ---- Horizontal 1x9 max, 4 outputs/thread via shared-window trick.
  // hmax[r][c..c+3] from tile[r][c..c+11]: 3x ds_load_b128, 13 v_max.
  // 72 rows x 16 col-groups = 1152 tasks.
  constexpr int NHT = IN_DIM * (TILE / 4);  // 1152
  for (int idx = tid; idx < NHT; idx += NTHREADS) {
    int r  = idx >> 4;          // 0..71
    int c4 = (idx & 15) << 2;   // 0,4,...,60
    const f32x4* p = (const f32x4*)&tile[r * IN_DIM + c4];  // 16B aligned
    f32x4 a = p[0], b = p[1], d = p[2];   // e0..e3, e4..e7, e8..e11

    // m = max(e3..e8)
    float m = fmaxf(fmaxf(fmaxf(a.w, b.x), fmaxf(b.y, b.z)), fmaxf(b.w, d.x));
    float p2 = fmaxf(a.z, m);     // max(e2..e8)
    float p1 = fmaxf(a.y, p2);    // max(e1..e8)
    float p0 = fmaxf(a.x, p1);    // max(e0..e8)
    float q  = fmaxf(d.y, d.z);   // max(e9,e10)
    float q2 = fmaxf(q, d.w);     // max(e9..e11)

    f32x4 o;
    o.x = p0;                     // max(e0..e8)
    o.y = fmaxf(p1, d.y);         // max(e1..e9)
    o.z = fmaxf(p2, q);           // max(e2..e10)
    o.w = fmaxf(m, q2);           // max(e3..e11)
    *(f32x4*)&hmax[r * TILE + c4] = o;
  }
  __syncthreads();

  // ---- Vertical 9x1 max + peak test, 4x4 output block per thread.
  // Exactly 16x16 = 256 blocks -> one per thread, no loop.
  {
    int r0 = (tid >> 4) << 2;   // 0,4,...,60
    int c0 = (tid & 15) << 2;   // 0,4,...,60

    const f32x4* col = (const f32x4*)&hmax[r0 * TILE + c0];  // stride 16 f32x4
    f32x4 v0  = col[0 * 16], v1  = col[1 * 16], v2  = col[2 * 16];
    f32x4 v3  = col[3 * 16], v4  = col[4 * 16], v5  = col[5 * 16];
    f32x4 v6  = col[6 * 16], v7  = col[7 * 16], v8  = col[8 * 16];
    f32x4 v9  = col[9 * 16], v10 = col[10 * 16], v11 = col[11 * 16];

    f32x4 mv = vmax4(vmax4(vmax4(v3, v4), vmax4(v5, v6)), vmax4(v7, v8));
    f32x4 p2 = vmax4(v2, mv);
    f32x4 p1 = vmax4(v1, p2);
    f32x4 p0 = vmax4(v0, p1);
    f32x4 q  = vmax4(v9, v10);
    f32x4 q2 = vmax4(q, v11);

    f32x4 M[4];
    M[0] = p0;               // rows r0..r0+8
    M[1] = vmax4(p1, v9);    // rows r0+1..r0+9
    M[2] = vmax4(p2, q);     // rows r0+2..r0+10
    M[3] = vmax4(mv, q2);    // rows r0+3..r0+11

    const size_t gbase = (size_t)(by * TILE + r0) * IMG_W + (bx * TILE + c0);
#pragma unroll
    for (int j = 0; j < 4; ++j) {
      // out = (M > 0.5 && M == center) ? M : 0 (threshold commutes with max)
      f32x4 cen = *(const f32x4*)&tile[(r0 + RAD + j) * IN_DIM + (c0 + RAD)];
      f32x4 o;
      o.x = (M[j].x > 0.5f && M[j].x == cen.x) ? M[j].x : 0.0f;
      o.y = (M[j].y > 0.5f && M[j].y == cen.y) ? M[j].y : 0.0f;
      o.z = (M[j].z > 0.5f && M[j].z == cen.z) ? M[j].z : 0.0f;
      o.w = (M[j].w > 0.5f && M[j].w == cen.w) ? M[j].w : 0.0f;
      *(f32x4*)&out[gbase + (size_t)j * IMG_W] = o;
    }
  }
}

extern "C" void kernel_launch(void* const* d_in, const int* in_sizes, int n_in,
                              void* d_out, int out_size, void* d_ws, size_t ws_size,
                              hipStream_t stream) {
  (void)in_sizes; (void)n_in; (void)d_ws; (void)ws_size; (void)out_size;
  const float* pred = (const float*)d_in[0];
  float* out = (float*)d_out;
  dim3 grid(IMG_W / TILE, IMG_H / TILE);  // 128 x 128
  dim3 block(NTHREADS);
  peak_pool_kernel<<<grid, block, 0, stream>>>(pred, out);
}